// MOLAttn_43147241456348
// MI455X (gfx1250) — compile-verified
//
#include <hip/hip_runtime.h>
#include <math.h>

typedef __attribute__((ext_vector_type(2))) float v2f;
typedef __attribute__((ext_vector_type(4))) float v4f;
typedef __attribute__((ext_vector_type(8))) float v8f;

#define B_  64
#define N_  2048
#define E_  512
#define D_  256
#define H_  256
#define K_  5
#define G3  768   // 3*H
#define NCHUNK 16 // n-chunks for the streaming reduction

// ---- workspace layout (floats) ----
#define WS_GI    0                       // 64*768
#define WS_GH    (WS_GI + B_*G3)         // 64*768
#define WS_HNEW  (WS_GH + B_*G3)         // 64*256
#define WS_HID   (WS_HNEW + B_*H_)       // 64*5*256
#define WS_WSUM  (WS_HID + B_*K_*H_)     // 64
#define WS_ARG   (WS_WSUM + B_)          // 64

__device__ __forceinline__ float sigmoidf_(float x) {
    return 1.0f / (1.0f + __expf(-x));
}

// ---------------------------------------------------------------------------
// K0: zero the c region of d_out (B*E floats). Needed because the final
// reduction accumulates with atomics and the harness poisons d_out.
// ---------------------------------------------------------------------------
__global__ void molattn_zero_c(float* __restrict__ out) {
    int idx = blockIdx.x * blockDim.x + threadIdx.x;
    if (idx < B_ * E_) out[idx] = 0.0f;
}

// ---------------------------------------------------------------------------
// K1: GI = dec_z @ W_ih[:, :256]^T   (64 x 768), GH = h0 @ W_hh^T (64 x 768)
// One wave (32 threads) per 16x16 output tile, v_wmma_f32_16x16x4_f32 over K=256.
// grid.x = 2 * 4 * 48 tiles.
// A-frag (16x4 f32): lanes 0-15 rows M, VGPR0/1 = K {0,1} (half 0) / {2,3} (half 1)
// B-frag (4x16):     lanes hold column N = lane&15, VGPR0/1 = K rows as above
// C/D:  lane l, vgpr v -> (M = v + 8*(l>>4), N = l&15)
// ---------------------------------------------------------------------------
__global__ void molattn_gru_gemm(const float* __restrict__ dec_z,
                                 const float* __restrict__ h0,
                                 const float* __restrict__ W_ih,
                                 const float* __restrict__ W_hh,
                                 float* __restrict__ ws) {
    const int tile  = blockIdx.x;          // 0 .. 383
    const int which = tile / (4 * 48);     // 0 = GI, 1 = GH
    const int rem   = tile % (4 * 48);
    const int mt    = rem / 48;            // M tile (rows of batch)
    const int nt    = rem % 48;            // N tile (gate columns)
    const int m0    = mt * 16;
    const int n0    = nt * 16;

    const float* Asrc = which ? h0 : dec_z;               // (64 x 256)
    const float* Wsrc = which ? W_hh : W_ih;              // row-major
    const int    ldw  = which ? H_ : G3;                  // W_hh: 256, W_ih: 768
    float*       Osrc = ws + (which ? WS_GH : WS_GI);

    const int lane = threadIdx.x;
    const int half = lane >> 4;       // 0 or 1
    const int lid  = lane & 15;

    v8f acc = {};
    const float* aRow = Asrc + (m0 + lid) * D_;           // row of A
    const float* bRow = Wsrc + (n0 + lid) * ldw;          // row g of W (gives B[k,g])

    #pragma unroll 4
    for (int k = 0; k < D_; k += 4) {
        const int kk = k + 2 * half;
        v2f a = *(const v2f*)(aRow + kk);
        v2f b = *(const v2f*)(bRow + kk);
        acc = __builtin_amdgcn_wmma_f32_16x16x4_f32(false, a, false, b,
                                                    (short)0, acc, false, false);
    }

    // store tile (bias added later in the gate kernel)
    #pragma unroll
    for (int v = 0; v < 8; ++v) {
        int m = m0 + v + 8 * half;
        int g = n0 + lid;
        Osrc[m * G3 + g] = acc[v];
    }
}

// ---------------------------------------------------------------------------
// K2: gate nonlinearity -> h_new.  h_new written to ws and to d_out (h region)
// ---------------------------------------------------------------------------
__global__ void molattn_gates(const float* __restrict__ h0,
                              const float* __restrict__ b_ih,
                              const float* __restrict__ b_hh,
                              float* __restrict__ ws,
                              float* __restrict__ out) {
    const int b = blockIdx.x;
    const int h = threadIdx.x;
    const float* gi = ws + WS_GI + b * G3;
    const float* gh = ws + WS_GH + b * G3;

    float gir = gi[h]            + b_ih[h];
    float ghr = gh[h]            + b_hh[h];
    float giz = gi[H_ + h]       + b_ih[H_ + h];
    float ghz = gh[H_ + h]       + b_hh[H_ + h];
    float gin = gi[2 * H_ + h]   + b_ih[2 * H_ + h];
    float ghn = gh[2 * H_ + h]   + b_hh[2 * H_ + h];

    float r  = sigmoidf_(gir + ghr);
    float z  = sigmoidf_(giz + ghz);
    float nn = tanhf(gin + r * ghn);
    float hp = h0[b * H_ + h];
    float hn = (1.0f - z) * nn + z * hp;

    ws[WS_HNEW + b * H_ + h] = hn;
    out[B_ * E_ + b * H_ + h] = hn;      // h_new is second output
}

// ---------------------------------------------------------------------------
// K3: hid[b,kc,h] = tanh( h_new(64x256) @ W1[kc]^T + b1[kc] )
// WMMA tiles: 5 comps * 4 Mtiles * 16 Ntiles = 320 waves.
// ---------------------------------------------------------------------------
__global__ void molattn_mlp1(const float* __restrict__ W1,
                             const float* __restrict__ b1,
                             float* __restrict__ ws) {
    const int tile = blockIdx.x;          // 0 .. 319
    const int kc   = tile / (4 * 16);
    const int rem  = tile % (4 * 16);
    const int mt   = rem / 16;
    const int nt   = rem % 16;
    const int m0   = mt * 16;
    const int n0   = nt * 16;

    const float* A  = ws + WS_HNEW;                  // (64 x 256)
    const float* Wk = W1 + kc * H_ * H_;             // (H x H): W1[kc,h,d]

    const int lane = threadIdx.x;
    const int half = lane >> 4;
    const int lid  = lane & 15;

    v8f acc = {};
    const float* aRow = A  + (m0 + lid) * H_;
    const float* bRow = Wk + (n0 + lid) * H_;        // B[k,n] = W1[kc,n,k]

    #pragma unroll 4
    for (int k = 0; k < H_; k += 4) {
        const int kk = k + 2 * half;
        v2f a = *(const v2f*)(aRow + kk);
        v2f b = *(const v2f*)(bRow + kk);
        acc = __builtin_amdgcn_wmma_f32_16x16x4_f32(false, a, false, b,
                                                    (short)0, acc, false, false);
    }

    #pragma unroll
    for (int v = 0; v < 8; ++v) {
        int m   = m0 + v + 8 * half;
        int col = n0 + lid;
        float hv = tanhf(acc[v] + b1[kc * H_ + col]);
        ws[WS_HID + m * (K_ * H_) + kc * H_ + col] = hv;   // (B, K, H)
    }
}

// ---------------------------------------------------------------------------
// K4: per-batch params (B x K x 3), softmax weight-sum and arg = mean/scale.
// One wave per batch; lane-strided dot + shfl reduction.
// ---------------------------------------------------------------------------
__global__ void molattn_params(const float* __restrict__ W2,
                               const float* __restrict__ b2,
                               float* __restrict__ ws) {
    const int b    = blockIdx.x;
    const int lane = threadIdx.x;        // 0..31
    float pr[15];

    #pragma unroll
    for (int p = 0; p < 15; ++p) {
        const int kc = p / 3, pp = p % 3;
        const float* hid = ws + WS_HID + b * (K_ * H_) + kc * H_;
        const float* w   = W2 + kc * (3 * H_) + pp * H_;
        float s = 0.0f;
        for (int h = lane; h < H_; h += 32) s += w[h] * hid[h];
        #pragma unroll
        for (int off = 16; off >= 1; off >>= 1) s += __shfl_down(s, off, 32);
        pr[p] = s + b2[kc * 3 + pp];     // valid in lane 0
    }

    if (lane == 0) {
        // softmax over logits pr[3k+2], then sum (faithful to reference)
        float l0 = pr[2], l1 = pr[5], l2 = pr[8], l3 = pr[11], l4 = pr[14];
        float mx = fmaxf(fmaxf(fmaxf(l0, l1), fmaxf(l2, l3)), l4);
        float e0 = __expf(l0 - mx), e1 = __expf(l1 - mx), e2 = __expf(l2 - mx),
              e3 = __expf(l3 - mx), e4 = __expf(l4 - mx);
        float S  = e0 + e1 + e2 + e3 + e4;
        float wsum = e0 / S + e1 / S + e2 / S + e3 / S + e4 / S;
        float mean  = __expf(pr[12]);    // params[:, K-1, 0]
        float scale = __expf(pr[13]);    // params[:, K-1, 1]
        ws[WS_WSUM + b] = wsum;
        ws[WS_ARG  + b] = mean / scale;
    }
}

// ---------------------------------------------------------------------------
// K5: c[b,e] += sum_n a[b,n] * enc_z[b,n,e]  -- the 256 MB streaming pass.
// grid = (B, NCHUNK=16); block = 128 threads, each owns 4 e columns (float4).
// enc_z is read exactly once -> non-temporal loads (TH=NT) to avoid L2 churn.
// a[b,n] = wsum * (sigmoid(n+0.5-arg) - sigmoid(n-0.5-arg)), staged in LDS.
// Partials combined via global f32 atomics into pre-zeroed c.
// ---------------------------------------------------------------------------
__global__ void molattn_reduce(const float* __restrict__ enc_z,
                               const float* __restrict__ ws,
                               float* __restrict__ out) {
    const int b     = blockIdx.x;
    const int chunk = blockIdx.y;
    const int tx    = threadIdx.x;       // 0..127
    const int nrows = N_ / NCHUNK;       // 128 rows per chunk

    __shared__ float a_s[N_ / NCHUNK];
    const float wsum = ws[WS_WSUM + b];
    const float arg  = ws[WS_ARG + b];

    {
        int i = tx;                      // 128 threads, 128 rows
        float t  = (float)(chunk * nrows + i);
        float f1 = sigmoidf_(t + 0.5f - arg);
        float f2 = sigmoidf_(t - 0.5f - arg);
        a_s[i] = wsum * (f1 - f2);
    }
    __syncthreads();

    const v4f* src =
        (const v4f*)(enc_z + ((size_t)b * N_ + (size_t)chunk * nrows) * E_) + tx;
    v4f acc = {};

    #pragma unroll 8
    for (int i = 0; i < nrows; ++i) {
        float a = a_s[i];
        v4f   v = __builtin_nontemporal_load(src + i * (E_ / 4));
        acc += a * v;
    }

    float* c = out + b * E_ + tx * 4;
    __hip_atomic_fetch_add(c + 0, acc.x, __ATOMIC_RELAXED, __HIP_MEMORY_SCOPE_AGENT);
    __hip_atomic_fetch_add(c + 1, acc.y, __ATOMIC_RELAXED, __HIP_MEMORY_SCOPE_AGENT);
    __hip_atomic_fetch_add(c + 2, acc.z, __ATOMIC_RELAXED, __HIP_MEMORY_SCOPE_AGENT);
    __hip_atomic_fetch_add(c + 3, acc.w, __ATOMIC_RELAXED, __HIP_MEMORY_SCOPE_AGENT);
}

// ---------------------------------------------------------------------------
extern "C" void kernel_launch(void* const* d_in, const int* in_sizes, int n_in,
                              void* d_out, int out_size, void* d_ws, size_t ws_size,
                              hipStream_t stream) {
    const float* enc_z = (const float*)d_in[0];
    const float* dec_z = (const float*)d_in[1];
    const float* h0    = (const float*)d_in[2];   // (1,B,H) -> flat B*H
    const float* W_ih  = (const float*)d_in[3];
    const float* W_hh  = (const float*)d_in[4];
    const float* b_ih  = (const float*)d_in[5];
    const float* b_hh  = (const float*)d_in[6];
    const float* W1    = (const float*)d_in[7];
    const float* b1    = (const float*)d_in[8];
    const float* W2    = (const float*)d_in[9];
    const float* b2    = (const float*)d_in[10];
    float* out = (float*)d_out;
    float* ws  = (float*)d_ws;

    // zero c region (atomically accumulated later)
    molattn_zero_c<<<(B_ * E_ + 255) / 256, 256, 0, stream>>>(out);

    // GI / GH via fp32 WMMA (384 one-wave tiles)
    molattn_gru_gemm<<<2 * 4 * 48, 32, 0, stream>>>(dec_z, h0, W_ih, W_hh, ws);

    // gate nonlinearity -> h_new
    molattn_gates<<<B_, H_, 0, stream>>>(h0, b_ih, b_hh, ws, out);

    // per-component MLP hidden via fp32 WMMA (320 one-wave tiles)
    molattn_mlp1<<<K_ * 4 * 16, 32, 0, stream>>>(W1, b1, ws);

    // per-batch params + softmax-sum + arg
    molattn_params<<<B_, 32, 0, stream>>>(W2, b2, ws);

    // 256 MB streaming reduction (the bandwidth-bound pass)
    dim3 grid5(B_, NCHUNK);
    molattn_reduce<<<grid5, 128, 0, stream>>>(enc_z, ws, out);
}